// GCNFlowArchEmbedder_32091995636316
// MI455X (gfx1250) — compile-verified
//
#include <hip/hip_runtime.h>
#include <math.h>

// ---------------------------------------------------------------------------
// GCN flow arch embedder for MI455X (gfx1250, wave32, WMMA).
// Heavy op: per-graph (4x128)@(128x128) fp32 GEMM -> batched as 16x128x128
// tiles (4 graphs per wave) using V_WMMA_F32_16X16X4_F32 (exact fp32 FMA).
// A panel lives fully in VGPRs (64 regs) and is reused across all 8 N-tiles.
// K loop is software-pipelined: B loads issued 4-deep, two accumulator chains.
// ---------------------------------------------------------------------------

typedef __attribute__((ext_vector_type(2))) float v2f;
typedef __attribute__((ext_vector_type(8))) float v8f;

#define NUM_GRAPHS 8192      // B * NCG
#define GPB 32               // graphs per block (8 waves * 4 graphs)
#define NBLOCKS (NUM_GRAPHS / GPB)

// workspace rows (packed stride 128): s1[12] | m1[7] | m2[7] | s2_01[4]
#define ROW_S1 0
#define ROW_M1 12
#define ROW_M2 19
#define ROW_S2 26
#define WS_FLOATS (30 * 128)

// main-kernel LDS layout (floats)
#define PAR_STRIDE 130       // padded row stride -> conflict-free row-indexed b64 loads
#define DT_STRIDE  20        // padded D-tile row stride -> half-waves on disjoint banks
#define LDS_PAR 16384        // after W2 [128*128]
#define LDS_DT  20288        // after 30*130 params (20284, aligned up)
#define LDS_ARCH 22848       // after 8 waves * 16*20 D-tile scratch
#define LDS_FLOATS 23360     // total: 93440 bytes

// ---------------------------------------------------------------------------
// Precompute (tiny, one block): everything that does not depend on per-graph
// edges. s1[c][i][:] = (node@Wx+bx)@W1 ; m{1,2}[op] = sigmoid(op@Wa+ba)*(op!=0)
// s2_01[c][i<2][:] = relu(s1[c][i]) @ W2  (nodes 0,1 have no incoming edges).
// ---------------------------------------------------------------------------
__global__ void gcn_precompute(const float* __restrict__ init_emb,
                               const float* __restrict__ op_embs,
                               const float* __restrict__ Wx,
                               const float* __restrict__ bx,
                               const float* __restrict__ W1,
                               const float* __restrict__ Wa1,
                               const float* __restrict__ ba1,
                               const float* __restrict__ W2,
                               const float* __restrict__ Wa2,
                               const float* __restrict__ ba2,
                               float* __restrict__ ws) {
  __shared__ float ysh[2 * 6 * 48];
  __shared__ float s1sh[2 * 6 * 128];
  const int t = threadIdx.x;

  // y = node @ Wx + bx   (nodes >= 2 are zero vectors -> y = bx)
  for (int idx = t; idx < 2 * 6 * 48; idx += 256) {
    int h = idx % 48, i = (idx / 48) % 6, c = idx / (48 * 6);
    float acc = bx[h];
    if (i < 2)
      for (int d = 0; d < 48; ++d)
        acc += init_emb[(c * 2 + i) * 48 + d] * Wx[d * 48 + h];
    ysh[idx] = acc;
  }
  __syncthreads();

  // s1 = y @ W1
  for (int idx = t; idx < 2 * 6 * 128; idx += 256) {
    int ch = idx % 128, i = (idx / 128) % 6, c = idx / (128 * 6);
    float acc = 0.f;
    for (int h = 0; h < 48; ++h)
      acc += ysh[(c * 6 + i) * 48 + h] * W1[h * 128 + ch];
    s1sh[idx] = acc;
    ws[ROW_S1 * 128 + idx] = acc;
  }
  __syncthreads();

  // s2_01 = relu(s1[:, :2]) @ W2   (layer-2 support rows for nodes 0,1)
  for (int idx = t; idx < 2 * 2 * 128; idx += 256) {
    int ch = idx % 128, i = (idx / 128) % 2, c = idx / 256;
    float acc = 0.f;
    for (int h = 0; h < 128; ++h)
      acc += fmaxf(0.f, s1sh[(c * 6 + i) * 128 + h]) * W2[h * 128 + ch];
    ws[ROW_S2 * 128 + idx] = acc;
  }

  // masked attention tables for both layers
  for (int idx = t; idx < 7 * 128; idx += 256) {
    int ch = idx % 128, op = idx / 128;
    float a1 = ba1[ch], a2 = ba2[ch];
    for (int d = 0; d < 48; ++d) {
      float e = op_embs[op * 48 + d];
      a1 += e * Wa1[d * 128 + ch];
      a2 += e * Wa2[d * 128 + ch];
    }
    float m = (op == 0) ? 0.f : 1.f;
    ws[ROW_M1 * 128 + idx] = m / (1.f + expf(-a1));
    ws[ROW_M2 * 128 + idx] = m / (1.f + expf(-a2));
  }
}

// ---------------------------------------------------------------------------
// Main kernel: per wave, 4 graphs (M=16 rows = 4 graphs x nodes 2..5).
//   h = relu(s1[i] + m1[opA] * s1[fA] + m1[opB] * s1[fB])   -> A fragments
//   support2 = h @ W2  via 8 N-tiles x 32 K-steps of wmma_f32_16x16x4_f32
//   out[g]   = 0.25 * sum_i (support2[i] + sum_k m2[op] * support2[f])
// A-fragment layout (ISA 32-bit A 16x4): lanes 0-15 hold K={0,1}, lanes 16-31
// hold K={2,3} of each 4-wide K chunk, one VGPR per K.
// ---------------------------------------------------------------------------
__global__ void __launch_bounds__(256)
gcn_main(const int* __restrict__ archs, const float* __restrict__ ws,
         const float* __restrict__ W2, float* __restrict__ out) {
  extern __shared__ float smem[];
  float* lds_w2 = smem;                        // [128][128]
  float* lds_par = smem + LDS_PAR;             // 30 rows, stride 130
  float* lds_dtb = smem + LDS_DT;              // 8 waves * [16][20]
  int* lds_arch = (int*)(smem + LDS_ARCH);     // 32 graphs * 16 ints

  const int tid = threadIdx.x;

  // Stage W2 (64 KB) into LDS, vectorized b128.
  for (int i = tid; i < 128 * 128 / 4; i += 256)
    ((float4*)lds_w2)[i] = ((const float4*)W2)[i];
  // Stage parameter tables with padded stride 130.
  for (int idx = tid; idx < WS_FLOATS; idx += 256) {
    int r = idx >> 7, ch = idx & 127;
    lds_par[r * PAR_STRIDE + ch] = ws[idx];
  }
  // Stage this block's 32 graph descriptors (16 ints each).
  {
    const int4* src = (const int4*)(archs + (size_t)blockIdx.x * GPB * 16);
    for (int i = tid; i < GPB * 16 / 4; i += 256) ((int4*)lds_arch)[i] = src[i];
  }
  __syncthreads();

  const int wave = tid >> 5, lane = tid & 31;
  const int row = lane & 15, hi = lane >> 4;   // M-row of A; K half select
  const int gq = row >> 2, node = row & 3;     // graph-in-wave, node-2
  const int gl = wave * 4 + gq;
  const int g = blockIdx.x * GPB + gl;
  const int c = g & 1;                         // NCG == 2

  const int* ga = lds_arch + gl * 16;
  const int fA = ga[2 * node], opA = ga[8 + 2 * node];
  const int fB = ga[2 * node + 1], opB = ga[8 + 2 * node + 1];

  const v2f* sv = (const v2f*)(lds_par + (ROW_S1 + c * 6 + 2 + node) * PAR_STRIDE);
  const v2f* sa = (const v2f*)(lds_par + (ROW_S1 + c * 6 + fA) * PAR_STRIDE);
  const v2f* sb = (const v2f*)(lds_par + (ROW_S1 + c * 6 + fB) * PAR_STRIDE);
  const v2f* ma = (const v2f*)(lds_par + (ROW_M1 + opA) * PAR_STRIDE);
  const v2f* mb = (const v2f*)(lds_par + (ROW_M1 + opB) * PAR_STRIDE);

  // Build the 16x128 A panel directly in fragment registers (32 K-steps x v2f).
  // FULLY unrolled so afrag[] has constant indices -> stays in VGPRs.
  v2f afrag[32];
#pragma unroll
  for (int kk = 0; kk < 32; ++kk) {
    int p = kk * 2 + hi;                       // float2 pair: ch = 4*kk + 2*hi
    v2f v = sv[p], a = sa[p], b = sb[p], x = ma[p], y = mb[p];
    v2f h;
    h.x = fmaxf(0.f, v.x + x.x * a.x + y.x * b.x);
    h.y = fmaxf(0.f, v.y + x.y * a.y + y.y * b.y);
    afrag[kk] = h;
  }

  float* dt = lds_dtb + wave * (16 * DT_STRIDE);
  const int qa = lane >> 3, ca = lane & 7;     // aggregation mapping
  const int* gaa = lds_arch + (wave * 4 + qa) * 16;
  const int gg = blockIdx.x * GPB + wave * 4 + qa;
  const int cc = gg & 1;

  // Hoist nt-invariant edge descriptors for the aggregation phase.
  int fagg[8], opagg[8];
#pragma unroll
  for (int e = 0; e < 8; ++e) {
    fagg[e] = gaa[e];
    opagg[e] = gaa[8 + e];
  }

  for (int nt = 0; nt < 8; ++nt) {
    // Two accumulator chains + grouped B loads -> 4 DS loads in flight and
    // WMMA issue overlapped with LDS latency instead of load/wait/mma lockstep.
    v8f acc0 = {}, acc1 = {};
    const int ncol = nt * 16 + row;            // B-fragment column
    const float* w2c = lds_w2 + (hi * 2) * 128 + ncol;
#pragma unroll
    for (int kq = 0; kq < 8; ++kq) {
      v2f bf0, bf1, bf2, bf3;
      const float* p0 = w2c + (kq * 16 + 0) * 128;
      const float* p1 = w2c + (kq * 16 + 4) * 128;
      const float* p2 = w2c + (kq * 16 + 8) * 128;
      const float* p3 = w2c + (kq * 16 + 12) * 128;
      bf0.x = p0[0]; bf0.y = p0[128];
      bf1.x = p1[0]; bf1.y = p1[128];
      bf2.x = p2[0]; bf2.y = p2[128];
      bf3.x = p3[0]; bf3.y = p3[128];
      acc0 = __builtin_amdgcn_wmma_f32_16x16x4_f32(false, afrag[kq * 4 + 0], false, bf0,
                                                   (short)0, acc0, false, false);
      acc1 = __builtin_amdgcn_wmma_f32_16x16x4_f32(false, afrag[kq * 4 + 1], false, bf1,
                                                   (short)0, acc1, false, false);
      acc0 = __builtin_amdgcn_wmma_f32_16x16x4_f32(false, afrag[kq * 4 + 2], false, bf2,
                                                   (short)0, acc0, false, false);
      acc1 = __builtin_amdgcn_wmma_f32_16x16x4_f32(false, afrag[kq * 4 + 3], false, bf3,
                                                   (short)0, acc1, false, false);
    }
    v8f acc = acc0 + acc1;

    // D layout: VGPR v -> M = v + 8*hi, N = row. Spill tile to padded LDS.
#pragma unroll
    for (int v = 0; v < 8; ++v) dt[(v + 8 * hi) * DT_STRIDE + row] = acc[v];

    // Same-wave LDS ops are in-order: stores above are visible to loads below.
    // Layer-2 edge aggregation + mean over nodes 2..5; 2 outputs per lane.
#pragma unroll
    for (int rep = 0; rep < 2; ++rep) {
      const int col = ca + rep * 8;
      const int n = nt * 16 + col;
      float o = 0.f;
#pragma unroll
      for (int i = 0; i < 4; ++i) {
        o += dt[(qa * 4 + i) * DT_STRIDE + col];
#pragma unroll
        for (int k = 0; k < 2; ++k) {
          const int f = fagg[2 * i + k];
          const int op = opagg[2 * i + k];
          const float sf = (f < 2)
              ? lds_par[(ROW_S2 + cc * 2 + f) * PAR_STRIDE + n]
              : dt[(qa * 4 + (f - 2)) * DT_STRIDE + col];
          o += lds_par[(ROW_M2 + op) * PAR_STRIDE + n] * sf;
        }
      }
      out[(size_t)gg * 128 + n] = 0.25f * o;
    }
  }
}

// ---------------------------------------------------------------------------
extern "C" void kernel_launch(void* const* d_in, const int* in_sizes, int n_in,
                              void* d_out, int out_size, void* d_ws,
                              size_t ws_size, hipStream_t stream) {
  const int* archs = (const int*)d_in[0];
  const float* init_emb = (const float*)d_in[1];
  // d_in[2] other_node_emb: all zeros, unused
  const float* op_embs = (const float*)d_in[3];
  const float* Wx = (const float*)d_in[4];
  const float* bx = (const float*)d_in[5];
  const float* W1 = (const float*)d_in[6];
  const float* Wa1 = (const float*)d_in[7];
  const float* ba1 = (const float*)d_in[8];
  const float* W2 = (const float*)d_in[9];
  const float* Wa2 = (const float*)d_in[10];
  const float* ba2 = (const float*)d_in[11];
  float* out = (float*)d_out;
  float* ws = (float*)d_ws;

  gcn_precompute<<<1, 256, 0, stream>>>(init_emb, op_embs, Wx, bx, W1, Wa1,
                                        ba1, W2, Wa2, ba2, ws);

  const size_t shmem = (size_t)LDS_FLOATS * sizeof(float);  // ~91 KB < 320 KB WGP
  (void)hipFuncSetAttribute((const void*)gcn_main,
                            hipFuncAttributeMaxDynamicSharedMemorySize,
                            (int)shmem);
  gcn_main<<<NBLOCKS, 256, shmem, stream>>>(archs, ws, W2, out);
}